// TemporalGATLayer_2259152798324
// MI455X (gfx1250) — compile-verified
//
#include <hip/hip_runtime.h>
#include <hip/hip_bf16.h>
#include <cstdint>

#define N_HEADS   4
#define D_HEAD    32
#define IN_DIM    128
#define TIME_DIM  128
#define K_TOT     256   // IN_DIM + TIME_DIM
#define OUT_DIM   128
#define NEG_SLOPE 0.2f

typedef __attribute__((ext_vector_type(2))) float    v2f;
typedef __attribute__((ext_vector_type(8))) float    v8f;
typedef __attribute__((ext_vector_type(4))) unsigned u32x4;
typedef __attribute__((ext_vector_type(8))) int      i32x8;
typedef __attribute__((ext_vector_type(4))) int      i32x4;

// monotone float <-> uint mapping so we can do float segment-max with u32 atomicMax
__device__ __forceinline__ unsigned enc_f32(float f) {
    unsigned u = __float_as_uint(f);
    return u ^ ((u >> 31) ? 0xFFFFFFFFu : 0x80000000u);
}
__device__ __forceinline__ float dec_f32(unsigned u) {
    unsigned b = (u >> 31) ? (u ^ 0x80000000u) : ~u;
    return __uint_as_float(b);
}
__device__ __forceinline__ float lrelu(float x) { return x > 0.f ? x : NEG_SLOPE * x; }

// ---------------------------------------------------------------------------
// K1: per-edge mailbox accumulation. One wave32 per edge; lane l handles
// dims [4l, 4l+4). sum_h[dst] += h[src]; sum_t[dst] += cos(t*w+b); deg[dst]++.
// ---------------------------------------------------------------------------
__global__ void k1_edge_accum(const float* __restrict__ h,
                              const float* __restrict__ time_t,
                              const int*   __restrict__ src,
                              const int*   __restrict__ dst,
                              const float* __restrict__ w_t,
                              const float* __restrict__ b_t,
                              float* __restrict__ deg,
                              float* __restrict__ sum_h,
                              float* __restrict__ sum_t,
                              int E) {
    int gtid = blockIdx.x * blockDim.x + threadIdx.x;
    int e    = gtid >> 5;
    int lane = gtid & 31;
    if (e >= E) return;
    int   s = src[e];
    int   d = dst[e];
    float t = time_t[e];
    int   k0 = lane * 4;

    float4 hv = *(const float4*)(h + (size_t)s * IN_DIM + k0);
    float* sh = sum_h + (size_t)d * IN_DIM + k0;
    atomicAdd(sh + 0, hv.x);
    atomicAdd(sh + 1, hv.y);
    atomicAdd(sh + 2, hv.z);
    atomicAdd(sh + 3, hv.w);

    float4 w4 = *(const float4*)(w_t + k0);
    float4 b4 = *(const float4*)(b_t + k0);
    float* st = sum_t + (size_t)d * TIME_DIM + k0;
    atomicAdd(st + 0, __cosf(t * w4.x + b4.x));
    atomicAdd(st + 1, __cosf(t * w4.y + b4.y));
    atomicAdd(st + 2, __cosf(t * w4.z + b4.z));
    atomicAdd(st + 3, __cosf(t * w4.w + b4.w));

    if (lane == 0) atomicAdd(deg + d, 1.0f);
}

// ---------------------------------------------------------------------------
// K2: feat = h_time @ W_fc.T and res = h_time @ W_res.T via fp32 WMMA with
// the weight matrix staged into LDS by the Tensor Data Mover.
//
// Block = 256 threads = 8 waves. blockIdx.y selects W_fc/W_res. Each wave
// issues one TENSOR_LOAD_TO_LDS for its 16-row slice of W (16 KB), waits on
// TENSORcnt, then the block barriers; B-fragments are then ds_load_b64 reads.
// Each wave computes one 16-row tile x 128 cols: 8 v8f accumulators,
// K-loop of 64 steps, 8 batched LDS loads then 8 WMMAs per step.
//
// A-frag (16x4 f32): lanes 0-15 hold K=kk,kk+1; lanes 16-31 hold K=kk+2,kk+3.
// B-frag (4x16 f32): B[k][n] = W[n][k]  (row-major W [128,256]).
// ---------------------------------------------------------------------------
__device__ __forceinline__ void tdm_stage_w_slice(const float* W, unsigned lds_off,
                                                  int slice_row0) {
    // 2D tile descriptor: tile_dim0 = 256 (contiguous K), tile_dim1 = 16 rows.
    unsigned long long gaddr =
        (unsigned long long)(uintptr_t)(W + (size_t)slice_row0 * K_TOT);

    u32x4 g0;
    g0.x = 1u;                                   // count=1, user mode, no gather
    g0.y = lds_off;                              // lds_addr (bytes)
    g0.z = (unsigned)(gaddr & 0xFFFFFFFFu);      // global_addr[31:0]
    g0.w = (unsigned)((gaddr >> 32) & 0x01FFFFFFu) | (2u << 30); // addr[56:32] | type=2

    const unsigned td0 = K_TOT;   // tensor_dim0 = 256 elements
    const unsigned td1 = 16;      // tensor_dim1 = 16 rows (this slice)
    const unsigned tile0 = K_TOT; // tile_dim0
    const unsigned tile1 = 16;    // tile_dim1
    const unsigned long long str0 = K_TOT; // tensor_dim0_stride (elements)

    i32x8 g1;
    g1[0] = (int)(2u << 16);                       // data_size=2 -> 4 bytes
    g1[1] = (int)((td0 & 0xFFFFu) << 16);          // tensor_dim0[15:0]
    g1[2] = (int)((td0 >> 16) | ((td1 & 0xFFFFu) << 16));   // td0[31:16] | td1[15:0]
    g1[3] = (int)((td1 >> 16) | (tile0 << 16));    // td1[31:16] | tile_dim0
    g1[4] = (int)(tile1);                          // tile_dim1 | tile_dim2=0
    g1[5] = (int)(unsigned)(str0 & 0xFFFFFFFFull); // tensor_dim0_stride[31:0]
    g1[6] = (int)(unsigned)((str0 >> 32) & 0xFFFFull); // stride0[47:32] | stride1[15:0]=0
    g1[7] = 0;                                     // stride1[47:16]=0

    i32x4 gz4 = {0, 0, 0, 0};                      // groups 2/3 unused (2D tensor)
    i32x8 gz8 = {0, 0, 0, 0, 0, 0, 0, 0};          // trailing group (clang-23 6-arg form)

    // amdgpu-toolchain (clang-23) arity: (u32x4, i32x8, i32x4, i32x4, i32x8, i32)
    __builtin_amdgcn_tensor_load_to_lds(g0, g1, gz4, gz4, gz8, 0);
    __builtin_amdgcn_s_wait_tensorcnt(0);
}

__global__ void k2_gemm_wmma(const float* __restrict__ deg,
                             const float* __restrict__ sum_h,
                             const float* __restrict__ sum_t,
                             const float* __restrict__ Wfc,
                             const float* __restrict__ Wres,
                             float* __restrict__ feat,
                             float* __restrict__ res,
                             int N) {
    extern __shared__ float ldsW[];   // [128][256] f32 = 128 KB (dynamic LDS)

    int lane   = threadIdx.x & 31;
    int waveId = __builtin_amdgcn_readfirstlane((int)(threadIdx.x >> 5));

    const float* W    = (blockIdx.y == 0) ? Wfc  : Wres;
    float*       outp = (blockIdx.y == 0) ? feat : res;

    // --- TDM stage: wave w copies W rows [16w, 16w+16) into LDS ---
    unsigned lds_base = (unsigned)(uintptr_t)&ldsW[0];
    tdm_stage_w_slice(W, lds_base + (unsigned)waveId * 16u * K_TOT * 4u, waveId * 16);
    __syncthreads();   // all slices resident before any wave reads them

    int rt   = blockIdx.x * (blockDim.x >> 5) + waveId;
    int row0 = rt * 16;
    if (row0 < N) {
        int half = lane >> 4;     // 0: K slots {0,1}, 1: K slots {2,3}
        int lm   = lane & 15;     // M (for A) / N (for B) within tile
        int mrow = row0 + lm; if (mrow > N - 1) mrow = N - 1;

        float invd = 1.0f / fmaxf(deg[mrow], 1.0f);
        const float* rowh = sum_h + (size_t)mrow * IN_DIM;
        const float* rowt = sum_t + (size_t)mrow * TIME_DIM;
        int koff = half * 2;

        v8f acc[8] = {};

        for (int phase = 0; phase < 2; ++phase) {
            const float* arow = phase ? rowt : rowh;
            int kbase = phase * IN_DIM;
            for (int kk = 0; kk < IN_DIM; kk += 4) {
                int k = kk + koff;
                v2f a;
                a.x = arow[k]     * invd;
                a.y = arow[k + 1] * invd;
                int kg = kbase + k;
                v2f b[8];
#pragma unroll
                for (int ct = 0; ct < 8; ++ct) {   // batched ds_load_b64
                    const float* bp = ldsW + (size_t)(ct * 16 + lm) * K_TOT + kg;
                    b[ct].x = bp[0];
                    b[ct].y = bp[1];
                }
#pragma unroll
                for (int ct = 0; ct < 8; ++ct) {
                    acc[ct] = __builtin_amdgcn_wmma_f32_16x16x4_f32(
                        /*neg_a=*/false, a, /*neg_b=*/false, b[ct],
                        /*c_mod=*/(short)0, acc[ct],
                        /*reuse_a=*/false, /*reuse_b=*/false);
                }
            }
        }

        // C/D layout: VGPR r, lanes 0-15 -> M=r, N=lane; lanes 16-31 -> M=r+8.
#pragma unroll
        for (int ct = 0; ct < 8; ++ct) {
#pragma unroll
            for (int r = 0; r < 8; ++r) {
                int row = row0 + r + half * 8;
                if (row < N) outp[(size_t)row * OUT_DIM + ct * 16 + lm] = acc[ct][r];
            }
        }
    }
}

// ---------------------------------------------------------------------------
// K3: per-(node,head) attention scores el/er (32-wide dot products).
// ---------------------------------------------------------------------------
__global__ void k3_scores(const float* __restrict__ feat,
                          const float* __restrict__ attn_l,
                          const float* __restrict__ attn_r,
                          float* __restrict__ el,
                          float* __restrict__ er,
                          int N) {
    int i = blockIdx.x * blockDim.x + threadIdx.x;
    if (i >= N * N_HEADS) return;
    int n  = i >> 2;
    int hh = i & 3;
    const float* f  = feat   + (size_t)n * OUT_DIM + hh * D_HEAD;
    const float* al = attn_l + hh * D_HEAD;
    const float* ar = attn_r + hh * D_HEAD;
    float l = 0.f, r = 0.f;
#pragma unroll
    for (int d = 0; d < D_HEAD; ++d) {
        l = fmaf(f[d], al[d], l);
        r = fmaf(f[d], ar[d], r);
    }
    el[i] = l;
    er[i] = r;
}

// ---------------------------------------------------------------------------
// K4: per-(edge,head) segment-max of leaky-relu scores (u32 atomicMax on
// monotone-encoded floats).
// ---------------------------------------------------------------------------
__global__ void k4_edge_max(const int* __restrict__ src,
                            const int* __restrict__ dst,
                            const float* __restrict__ el,
                            const float* __restrict__ er,
                            unsigned* __restrict__ m_enc,
                            int E) {
    int i = blockIdx.x * blockDim.x + threadIdx.x;
    if (i >= E * N_HEADS) return;
    int e  = i >> 2;
    int hh = i & 3;
    int s = src[e], d = dst[e];
    float x = lrelu(el[s * N_HEADS + hh] + er[d * N_HEADS + hh]);
    atomicMax(m_enc + d * N_HEADS + hh, enc_f32(x));
}

// ---------------------------------------------------------------------------
// K5: per-(edge,head) softmax denominator (recompute e; no [E,H] scratch).
// ---------------------------------------------------------------------------
__global__ void k5_edge_denom(const int* __restrict__ src,
                              const int* __restrict__ dst,
                              const float* __restrict__ el,
                              const float* __restrict__ er,
                              const unsigned* __restrict__ m_enc,
                              float* __restrict__ denom,
                              int E) {
    int i = blockIdx.x * blockDim.x + threadIdx.x;
    if (i >= E * N_HEADS) return;
    int e  = i >> 2;
    int hh = i & 3;
    int s = src[e], d = dst[e];
    int idx = d * N_HEADS + hh;
    float x = lrelu(el[s * N_HEADS + hh] + er[idx]);
    float m = dec_f32(m_enc[idx]);
    atomicAdd(denom + idx, __expf(x - m));
}

// ---------------------------------------------------------------------------
// K6: per-edge weighted aggregation: rst[dst] += a * feat[src].
// One wave32 per edge; lane l handles dims [4l,4l+4) -> head = l/8.
// rst lives in d_out (zero-initialized).
// ---------------------------------------------------------------------------
__global__ void k6_edge_agg(const int* __restrict__ src,
                            const int* __restrict__ dst,
                            const float* __restrict__ el,
                            const float* __restrict__ er,
                            const unsigned* __restrict__ m_enc,
                            const float* __restrict__ denom,
                            const float* __restrict__ feat,
                            float* __restrict__ rst,
                            int E) {
    int gtid = blockIdx.x * blockDim.x + threadIdx.x;
    int e    = gtid >> 5;
    int lane = gtid & 31;
    if (e >= E) return;
    int s  = src[e];
    int d  = dst[e];
    int hh = lane >> 3;
    int idx = d * N_HEADS + hh;

    float x = lrelu(el[s * N_HEADS + hh] + er[idx]);
    float m = dec_f32(m_enc[idx]);
    float a = __expf(x - m) / fmaxf(denom[idx], 1e-9f);

    int k0 = lane * 4;
    float4 f = *(const float4*)(feat + (size_t)s * OUT_DIM + k0);
    float* o = rst + (size_t)d * OUT_DIM + k0;
    atomicAdd(o + 0, a * f.x);
    atomicAdd(o + 1, a * f.y);
    atomicAdd(o + 2, a * f.z);
    atomicAdd(o + 3, a * f.w);
}

// ---------------------------------------------------------------------------
// K7: out = elu(rst + res + bias), in place on d_out.
// ---------------------------------------------------------------------------
__global__ void k7_final(const float* __restrict__ res,
                         const float* __restrict__ bias,
                         float* __restrict__ out,
                         int total) {
    int i = blockIdx.x * blockDim.x + threadIdx.x;
    if (i >= total) return;
    float v = out[i] + res[i] + bias[i & (OUT_DIM - 1)];
    out[i] = (v > 0.f) ? v : expm1f(v);
}

// ---------------------------------------------------------------------------
extern "C" void kernel_launch(void* const* d_in, const int* in_sizes, int n_in,
                              void* d_out, int out_size, void* d_ws, size_t ws_size,
                              hipStream_t stream) {
    const float* h      = (const float*)d_in[0];
    const float* time_t = (const float*)d_in[1];
    const int*   src    = (const int*)  d_in[2];
    const int*   dst    = (const int*)  d_in[3];
    // d_in[4] = num_nodes scalar (device side, unusable for grid dims)
    const float* w_t    = (const float*)d_in[5];
    const float* b_t    = (const float*)d_in[6];
    const float* Wfc    = (const float*)d_in[7];
    const float* attn_l = (const float*)d_in[8];
    const float* attn_r = (const float*)d_in[9];
    const float* Wres   = (const float*)d_in[10];
    const float* bias   = (const float*)d_in[11];

    int N = in_sizes[0] / IN_DIM;   // h is [N, 128]
    int E = in_sizes[1];            // time_tensor is [E]

    // workspace layout (all fp32-sized slots)
    float* ws = (float*)d_ws;
    size_t off = 0;
    float*    deg   = ws + off; off += (size_t)N;
    float*    sum_h = ws + off; off += (size_t)N * IN_DIM;
    float*    sum_t = ws + off; off += (size_t)N * TIME_DIM;
    float*    feat  = ws + off; off += (size_t)N * OUT_DIM;
    float*    res   = ws + off; off += (size_t)N * OUT_DIM;
    float*    el    = ws + off; off += (size_t)N * N_HEADS;
    float*    er    = ws + off; off += (size_t)N * N_HEADS;
    unsigned* m_enc = (unsigned*)(ws + off); off += (size_t)N * N_HEADS;
    float*    denom = ws + off; off += (size_t)N * N_HEADS;

    // zero-fill accumulators (capture-safe stream memsets); enc_f32 sentinel 0
    // for m_enc marks "no incoming edge" and is never decoded for such nodes.
    (void)hipMemsetAsync(d_ws, 0, off * sizeof(float), stream);
    (void)hipMemsetAsync(d_out, 0, (size_t)out_size * sizeof(float), stream);

    const int BLK = 256;

    // K1: one wave per edge
    {
        long long threads = (long long)E * 32;
        int blocks = (int)((threads + BLK - 1) / BLK);
        k1_edge_accum<<<blocks, BLK, 0, stream>>>(h, time_t, src, dst, w_t, b_t,
                                                  deg, sum_h, sum_t, E);
    }
    // K2: TDM-staged WMMA GEMMs (8 waves/block; y=0 -> W_fc/feat, y=1 -> W_res/res)
    {
        int numRT = (N + 15) / 16;
        dim3 grid((numRT + 7) / 8, 2);
        size_t ldsBytes = (size_t)128 * K_TOT * sizeof(float);  // 128 KB
        k2_gemm_wmma<<<grid, BLK, ldsBytes, stream>>>(deg, sum_h, sum_t, Wfc, Wres,
                                                      feat, res, N);
    }
    // K3: attention scores
    {
        int total = N * N_HEADS;
        k3_scores<<<(total + BLK - 1) / BLK, BLK, 0, stream>>>(feat, attn_l, attn_r,
                                                               el, er, N);
    }
    // K4: segment max
    {
        int total = E * N_HEADS;
        k4_edge_max<<<(total + BLK - 1) / BLK, BLK, 0, stream>>>(src, dst, el, er,
                                                                 m_enc, E);
    }
    // K5: softmax denominator
    {
        int total = E * N_HEADS;
        k5_edge_denom<<<(total + BLK - 1) / BLK, BLK, 0, stream>>>(src, dst, el, er,
                                                                   m_enc, denom, E);
    }
    // K6: weighted aggregation into d_out
    {
        long long threads = (long long)E * 32;
        int blocks = (int)((threads + BLK - 1) / BLK);
        k6_edge_agg<<<blocks, BLK, 0, stream>>>(src, dst, el, er, m_enc, denom,
                                                feat, (float*)d_out, E);
    }
    // K7: residual + bias + elu
    {
        int total = N * OUT_DIM;
        k7_final<<<(total + BLK - 1) / BLK, BLK, 0, stream>>>(res, bias,
                                                              (float*)d_out, total);
    }
}